// HiPPO_76699525972314
// MI455X (gfx1250) — compile-verified
//
#include <hip/hip_runtime.h>
#include <hip/hip_bf16.h>

typedef __attribute__((ext_vector_type(2))) float v2f;
typedef __attribute__((ext_vector_type(4))) float v4f;
typedef __attribute__((ext_vector_type(8))) float v8f;

#define NST   512            // state dimension
#define LSEQ  65536          // sequence length
#define CHUNK 32             // timesteps per level-0 chunk
#define M0    (LSEQ / CHUNK) // 2048 chunks
#define GRP   32             // chunks per level-1 group
#define NGRP  (M0 / GRP)     // 64 groups
#define COLS  32             // state columns per workgroup
#define SSTR  516            // padded LDS column stride (floats)
#define NW    8              // wave32s per workgroup
#define NTHR  (NW * 32)

__device__ __forceinline__ v8f wmma_f32(v2f a, v2f b, v8f c) {
    // V_WMMA_F32_16X16X4_F32: D = A(16x4) * B(4x16) + C(16x16), all f32
    return __builtin_amdgcn_wmma_f32_16x16x4_f32(false, a, false, b,
                                                 (short)0, c, false, false);
}

// ---------------------------------------------------------------------------
// Dense 512x512x512 f32 GEMM (used to build Ab^32 and Ab^1024 by squaring).
// One 16x16 tile per wave; 8 waves / WG; 128 WGs cover the 32x32 tile grid.
// ---------------------------------------------------------------------------
__global__ __launch_bounds__(NTHR) void gemm512_kernel(
    const float* __restrict__ A, const float* __restrict__ B,
    float* __restrict__ C) {
    const int lane  = threadIdx.x & 31;
    const int wv    = threadIdx.x >> 5;
    const int laneN = lane & 15;
    const int kOff  = (lane >> 4) << 1;   // lanes 16-31 hold K=2,3
    const int mOff  = (lane >> 4) << 3;   // C/D: lanes 16-31 hold M=r+8
    const int tile  = blockIdx.x * NW + wv;
    const int tm = tile >> 5, tn = tile & 31;

    v8f acc = {};
    const float* arow = A + (tm * 16 + laneN) * NST + kOff;
    const float* bcol = B + tn * 16 + laneN;
    #pragma unroll 4
    for (int kb = 0; kb < NST; kb += 4) {
        v2f af = *(const v2f*)(arow + kb);
        v2f bf;
        bf.x = bcol[(kb + kOff) * NST];
        bf.y = bcol[(kb + kOff + 1) * NST];
        acc = wmma_f32(af, bf, acc);
    }
    float* crow = C + (tm * 16 + mOff) * NST + tn * 16 + laneN;
    #pragma unroll
    for (int r = 0; r < 8; r++) crow[r * NST] = acc[r];
}

// ---------------------------------------------------------------------------
// Batched recurrence pass: 32 sequential steps of  X <- A*X + input_t
// over COLS columns per workgroup, X double-buffered in LDS.
//   MODE 0: chunk-local finals     (init 0,        input b*u,    write F)
//   MODE 1: final replay           (init carry1,   input b*u,    write out/step)
//   MODE 2: group-local finals     (init 0,        input F col,  write G)
//   MODE 3: carry replay           (init carry2,   input F col,  write carry1/step)
// ---------------------------------------------------------------------------
template <int MODE>
__global__ __launch_bounds__(NTHR) void pass_kernel(
    const float* __restrict__ A,      // Ab (MODE 0/1) or P=Ab^32 (MODE 2/3)
    const float* __restrict__ bvec,   // b (MODE 0/1)
    const float* __restrict__ u,      // u (MODE 0/1)
    const float* __restrict__ Fin,    // F (MODE 2/3)
    const float* __restrict__ carryIn,// carry1 (MODE 1) / carry2 (MODE 3)
    float* __restrict__ stepOut,      // out (MODE 1) / carry1 (MODE 3)
    float* __restrict__ finalOut) {   // F (MODE 0) / G (MODE 2)
    extern __shared__ float lds[];
    float* Sb0  = lds;
    float* Sb1  = lds + COLS * SSTR;
    float* bLds = lds + 2 * COLS * SSTR;

    const int tid   = threadIdx.x;
    const int lane  = tid & 31;
    const int wv    = tid >> 5;
    const int laneN = lane & 15;
    const int kOff  = (lane >> 4) << 1;
    const int mOff  = (lane >> 4) << 3;
    const int colBase = blockIdx.x * COLS;

    __builtin_prefetch(A + tid * 64, 0, 1);  // global_prefetch_b8: warm L2 with A

    if constexpr (MODE == 0 || MODE == 1) {
        for (int i = tid; i < NST; i += NTHR) bLds[i] = bvec[i];
    }
    // init state columns (column-major in LDS: S[col*SSTR + n])
    for (int idx = tid; idx < COLS * NST; idx += NTHR) {
        const int col = idx >> 9, n = idx & (NST - 1);
        float v = 0.f;
        if constexpr (MODE == 1 || MODE == 3)
            v = carryIn[(size_t)(colBase + col) * NST + n];
        Sb0[col * SSTR + n] = v;
    }
    __syncthreads();

    for (int t = 0; t < CHUNK; t++) {
        float* So = (t & 1) ? Sb1 : Sb0;
        float* Sn = (t & 1) ? Sb0 : Sb1;

        if constexpr (MODE == 3) {
            // carry1[g*GRP + t] = state BEFORE consuming chunk g*GRP+t
            for (int idx = tid; idx < COLS * NST; idx += NTHR) {
                const int col = idx >> 9, n = idx & (NST - 1);
                __builtin_nontemporal_store(
                    So[col * SSTR + n],
                    stepOut + ((size_t)((colBase + col) * GRP + t)) * NST + n);
            }
        }

        v8f acc[4][2] = {};
        #pragma unroll 2
        for (int kb = 0; kb < NST; kb += 4) {
            v2f af[4];
            #pragma unroll
            for (int i = 0; i < 4; i++) {
                const int row = (wv * 4 + i) * 16 + laneN;
                af[i] = *(const v2f*)(A + row * NST + kb + kOff);
            }
            v2f bf[2];
            #pragma unroll
            for (int j = 0; j < 2; j++) {
                const int col = j * 16 + laneN;
                bf[j] = *(const v2f*)(So + col * SSTR + kb + kOff);
            }
            #pragma unroll
            for (int i = 0; i < 4; i++)
                #pragma unroll
                for (int j = 0; j < 2; j++)
                    acc[i][j] = wmma_f32(af[i], bf[j], acc[i][j]);
        }

        // inject input for this step
        #pragma unroll
        for (int j = 0; j < 2; j++) {
            const int col = colBase + j * 16 + laneN;
            if constexpr (MODE == 0 || MODE == 1) {
                const float uv = u[(size_t)col * CHUNK + t];
                #pragma unroll
                for (int i = 0; i < 4; i++)
                    #pragma unroll
                    for (int r = 0; r < 8; r++)
                        acc[i][j][r] =
                            fmaf(bLds[(wv * 4 + i) * 16 + mOff + r], uv, acc[i][j][r]);
            } else {
                const float* fc = Fin + ((size_t)(col * GRP + t)) * NST;
                #pragma unroll
                for (int i = 0; i < 4; i++)
                    #pragma unroll
                    for (int r = 0; r < 8; r++)
                        acc[i][j][r] += fc[(wv * 4 + i) * 16 + mOff + r];
            }
        }

        // write new state into the other LDS buffer (and outputs for MODE 1)
        #pragma unroll
        for (int i = 0; i < 4; i++) {
            #pragma unroll
            for (int j = 0; j < 2; j++) {
                const int c16 = j * 16 + laneN;
                const int m0  = (wv * 4 + i) * 16 + mOff;
                v4f lo = { acc[i][j][0], acc[i][j][1], acc[i][j][2], acc[i][j][3] };
                v4f hi = { acc[i][j][4], acc[i][j][5], acc[i][j][6], acc[i][j][7] };
                *(v4f*)(Sn + c16 * SSTR + m0)     = lo;
                *(v4f*)(Sn + c16 * SSTR + m0 + 4) = hi;
                if constexpr (MODE == 1) {
                    const size_t orow =
                        ((size_t)(colBase + c16) * CHUNK + t) * NST + m0;
                    __builtin_nontemporal_store(lo, (v4f*)(stepOut + orow));
                    __builtin_nontemporal_store(hi, (v4f*)(stepOut + orow + 4));
                }
            }
        }
        __syncthreads();
    }

    if constexpr (MODE == 0 || MODE == 2) {
        float* Sf = (CHUNK & 1) ? Sb1 : Sb0;
        for (int idx = tid; idx < COLS * NST; idx += NTHR) {
            const int col = idx >> 9, n = idx & (NST - 1);
            __builtin_nontemporal_store(
                Sf[col * SSTR + n],
                finalOut + (size_t)(colBase + col) * NST + n);
        }
    }
}

// ---------------------------------------------------------------------------
// Top-level 64-step serial scan with P2 = Ab^1024 (tiny): carry2_g = x_{1024*g}
// ---------------------------------------------------------------------------
__global__ __launch_bounds__(NST) void scan_top_kernel(
    const float* __restrict__ P2, const float* __restrict__ G,
    const float* __restrict__ x0, float* __restrict__ carry2) {
    __shared__ float xs[NST];
    const int tid = threadIdx.x;
    xs[tid] = x0[tid];
    __syncthreads();
    for (int g = 0; g < NGRP; g++) {
        carry2[g * NST + tid] = xs[tid];
        float y = G[g * NST + tid];
        const float* prow = P2 + (size_t)tid * NST;
        #pragma unroll 8
        for (int k = 0; k < NST; k++) y = fmaf(prow[k], xs[k], y);
        __syncthreads();
        xs[tid] = y;
        __syncthreads();
    }
}

// ---------------------------------------------------------------------------
extern "C" void kernel_launch(void* const* d_in, const int* in_sizes, int n_in,
                              void* d_out, int out_size, void* d_ws, size_t ws_size,
                              hipStream_t stream) {
    const float* u  = (const float*)d_in[0];   // (65536,)
    const float* Ab = (const float*)d_in[1];   // (512,512) row-major
    const float* Bb = (const float*)d_in[2];   // (512,1) -> b contiguous
    const float* x0 = (const float*)d_in[3];   // (512,)
    float* out = (float*)d_out;                // (65536,512)

    float* ws = (float*)d_ws;                  // needs ~12.3 MB of scratch
    float* T1 = ws;
    float* T2 = T1 + NST * NST;
    float* P  = T2 + NST * NST;                // Ab^32
    float* P2 = P  + NST * NST;                // Ab^1024
    float* F  = P2 + NST * NST;                // (M0, 512) chunk-local finals
    float* G  = F  + (size_t)M0 * NST;         // (64, 512) group-local finals
    float* c2 = G  + (size_t)NGRP * NST;       // (64, 512) group carries
    float* c1 = c2 + (size_t)NGRP * NST;       // (M0, 512) chunk carries

    const size_t ldsBytes = (size_t)(2 * COLS * SSTR + NST) * sizeof(float);

    // P = Ab^32 via 5 squarings
    gemm512_kernel<<<128, NTHR, 0, stream>>>(Ab, Ab, T1);
    gemm512_kernel<<<128, NTHR, 0, stream>>>(T1, T1, T2);
    gemm512_kernel<<<128, NTHR, 0, stream>>>(T2, T2, T1);
    gemm512_kernel<<<128, NTHR, 0, stream>>>(T1, T1, T2);
    gemm512_kernel<<<128, NTHR, 0, stream>>>(T2, T2, P);
    // P2 = P^32 = Ab^1024
    gemm512_kernel<<<128, NTHR, 0, stream>>>(P, P, T1);
    gemm512_kernel<<<128, NTHR, 0, stream>>>(T1, T1, T2);
    gemm512_kernel<<<128, NTHR, 0, stream>>>(T2, T2, T1);
    gemm512_kernel<<<128, NTHR, 0, stream>>>(T1, T1, T2);
    gemm512_kernel<<<128, NTHR, 0, stream>>>(T2, T2, P2);

    // Pass A: chunk-local finals F (2048 chunks, zero-start)
    pass_kernel<0><<<M0 / COLS, NTHR, ldsBytes, stream>>>(
        Ab, Bb, u, nullptr, nullptr, nullptr, F);
    // Pass A2: group-local finals G (64 groups over F, with P)
    pass_kernel<2><<<NGRP / COLS, NTHR, ldsBytes, stream>>>(
        P, nullptr, nullptr, F, nullptr, nullptr, G);
    // Top scan: carry2 (serial, tiny)
    scan_top_kernel<<<1, NST, 0, stream>>>(P2, G, x0, c2);
    // Pass B2: regenerate per-chunk carries carry1
    pass_kernel<3><<<NGRP / COLS, NTHR, ldsBytes, stream>>>(
        P, nullptr, nullptr, F, c2, c1, nullptr);
    // Pass B: replay chunks seeded with carry1, write all 65536 states
    pass_kernel<1><<<M0 / COLS, NTHR, ldsBytes, stream>>>(
        Ab, Bb, u, nullptr, c1, out, nullptr);
}